// GCNClassifier_88648124990396
// MI455X (gfx1250) — compile-verified
//
#include <hip/hip_runtime.h>
#include <hip/hip_bf16.h>

#define EMB 128
#define NCLS 10

typedef __attribute__((ext_vector_type(2))) float v2f;
typedef __attribute__((ext_vector_type(8))) float v8f;

// ---------------------------------------------------------------------------
// Embedding: h[n][c] = shape_tab[x[n][0]][c] + color_tab[x[n][1]][c] + pos_tab[x[n][2]][c]
// One thread per float4 (N*32 threads).
__global__ void k_embed(const int* __restrict__ x,
                        const float* __restrict__ st,
                        const float* __restrict__ ct,
                        const float* __restrict__ pt,
                        float* __restrict__ h, int n)
{
    int idx = blockIdx.x * blockDim.x + threadIdx.x;
    if (idx >= n * 32) return;
    int node = idx >> 5;
    int q    = idx & 31;
    int s = x[node * 3 + 0];
    int c = x[node * 3 + 1];
    int p = x[node * 3 + 2];
    const float4 a = reinterpret_cast<const float4*>(st)[s * 32 + q];
    const float4 b = reinterpret_cast<const float4*>(ct)[c * 32 + q];
    const float4 d = reinterpret_cast<const float4*>(pt)[p * 32 + q];
    float4 o;
    o.x = a.x + b.x + d.x;
    o.y = a.y + b.y + d.y;
    o.z = a.z + b.z + d.z;
    o.w = a.w + b.w + d.w;
    reinterpret_cast<float4*>(h)[node * 32 + q] = o;
}

// ---------------------------------------------------------------------------
__global__ void k_fill(float* __restrict__ p, float v, int n)
{
    int idx = blockIdx.x * blockDim.x + threadIdx.x;
    for (; idx < n; idx += gridDim.x * blockDim.x) p[idx] = v;
}

__global__ void k_deg_accum(const int* __restrict__ dst, float* __restrict__ deg, int e)
{
    int idx = blockIdx.x * blockDim.x + threadIdx.x;
    if (idx >= e) return;
    unsafeAtomicAdd(&deg[dst[idx]], 1.0f);
}

__global__ void k_rsqrt_inplace(float* __restrict__ d, int n)
{
    int idx = blockIdx.x * blockDim.x + threadIdx.x;
    if (idx >= n) return;
    d[idx] = __frsqrt_rn(d[idx]);
}

// ---------------------------------------------------------------------------
// xw = h @ W   (nrows x 128) @ (128 x 128), fp32 WMMA 16x16x4.
// 256 threads = 8 waves; block covers 128 rows; W staged in LDS (64 KB).
#define WSTRIDE 128
__global__ __launch_bounds__(256) void k_gemm_wmma(
    const float* __restrict__ A, const float* __restrict__ W,
    float* __restrict__ C, int nrows)
{
    __shared__ float Wl[128 * WSTRIDE];
    const int tid = threadIdx.x;

    // Cooperative load of W (128x128 f32) into LDS, float4 granularity.
    for (int idx = tid; idx < 128 * 32; idx += 256) {
        int r = idx >> 5, q = idx & 31;
        reinterpret_cast<float4*>(&Wl[r * WSTRIDE + q * 4])[0] =
            reinterpret_cast<const float4*>(W)[r * 32 + q];
    }
    __syncthreads();

    const int wave  = tid >> 5;
    const int lane  = tid & 31;
    const int m     = lane & 15;        // row within 16-row tile (A), col within 16 (B/D)
    const int khalf = (lane >> 4) * 2;  // K sub-offset: lanes 0-15 -> K0/K1, 16-31 -> K2/K3
    const int tileRow = blockIdx.x * 128 + wave * 16;
    if (tileRow >= nrows) return;       // wave-uniform (nrows % 16 == 0): EXEC all-ones below

    v8f acc[8] = {};

    const float* arow = A + (long long)(tileRow + m) * EMB;
    for (int k = 0; k < EMB; k += 4) {
        const int kk = k + khalf;
        v2f a;
        a.x = arow[kk];
        a.y = arow[kk + 1];
#pragma unroll
        for (int j = 0; j < 8; ++j) {
            const int col = j * 16 + m;
            v2f b;
            b.x = Wl[kk * WSTRIDE + col];
            b.y = Wl[(kk + 1) * WSTRIDE + col];
            acc[j] = __builtin_amdgcn_wmma_f32_16x16x4_f32(
                false, a, false, b, (short)0, acc[j], false, false);
        }
    }

    // D layout: VGPR r -> M = r + 8*(lane>=16), N = lane&15
    const int rbase = (lane >> 4) * 8;
#pragma unroll
    for (int j = 0; j < 8; ++j) {
#pragma unroll
        for (int r = 0; r < 8; ++r) {
            C[(long long)(tileRow + rbase + r) * EMB + j * 16 + m] = acc[j][r];
        }
    }
}

// ---------------------------------------------------------------------------
// Edge aggregation: agg[dst] += xw[src] * dinv[src]*dinv[dst].  32 lanes/edge.
__global__ void k_agg_edges(const float* __restrict__ xw,
                            const int* __restrict__ src,
                            const int* __restrict__ dst,
                            const float* __restrict__ dinv,
                            float* __restrict__ agg, int e)
{
    int idx = blockIdx.x * blockDim.x + threadIdx.x;
    if (idx >= e * 32) return;
    const int ed   = idx >> 5;
    const int lane = idx & 31;
    const int s = src[ed];
    const int d = dst[ed];
    const float coef = dinv[s] * dinv[d];
    if (lane == 0 && ed + 64 < e) {
        // warm L2/L0 for an upcoming source row (speculative prefetch)
        __builtin_prefetch(xw + (long long)src[ed + 64] * EMB, 0, 0);
    }
    const float4 v = reinterpret_cast<const float4*>(xw)[(long long)s * 32 + lane];
    float* ap = agg + (long long)d * EMB + lane * 4;
    unsafeAtomicAdd(ap + 0, v.x * coef);
    unsafeAtomicAdd(ap + 1, v.y * coef);
    unsafeAtomicAdd(ap + 2, v.z * coef);
    unsafeAtomicAdd(ap + 3, v.w * coef);
}

// ---------------------------------------------------------------------------
// h = relu(agg + xw * dinv^2 + bias)
__global__ void k_finalize_relu(const float* __restrict__ agg,
                                const float* __restrict__ xw,
                                const float* __restrict__ dinv,
                                const float* __restrict__ bias,
                                float* __restrict__ h, int n)
{
    int idx = blockIdx.x * blockDim.x + threadIdx.x;
    if (idx >= n * 32) return;
    const int node = idx >> 5;
    const int q    = idx & 31;
    const float di = dinv[node];
    const float sl = di * di;
    const float4 ag = reinterpret_cast<const float4*>(agg)[(long long)node * 32 + q];
    const float4 xv = reinterpret_cast<const float4*>(xw)[(long long)node * 32 + q];
    const float* bp = bias + q * 4;
    float4 o;
    o.x = fmaxf(ag.x + xv.x * sl + bp[0], 0.0f);
    o.y = fmaxf(ag.y + xv.y * sl + bp[1], 0.0f);
    o.z = fmaxf(ag.z + xv.z * sl + bp[2], 0.0f);
    o.w = fmaxf(ag.w + xv.w * sl + bp[3], 0.0f);
    reinterpret_cast<float4*>(h)[(long long)node * 32 + q] = o;
}

// ---------------------------------------------------------------------------
// Mean-pool accumulation: sums[batch[n]] += h[n]; cnt[batch[n]] += 1
__global__ void k_pool(const float* __restrict__ h,
                       const int* __restrict__ batch,
                       float* __restrict__ sums,
                       float* __restrict__ cnt, int n)
{
    int idx = blockIdx.x * blockDim.x + threadIdx.x;
    if (idx >= n * 32) return;
    const int node = idx >> 5;
    const int q    = idx & 31;
    const int g = batch[node];
    const float4 v = reinterpret_cast<const float4*>(h)[(long long)node * 32 + q];
    float* sp = sums + (long long)g * EMB + q * 4;
    unsafeAtomicAdd(sp + 0, v.x);
    unsafeAtomicAdd(sp + 1, v.y);
    unsafeAtomicAdd(sp + 2, v.z);
    unsafeAtomicAdd(sp + 3, v.w);
    if (q == 0) unsafeAtomicAdd(&cnt[g], 1.0f);
}

// ---------------------------------------------------------------------------
// out[g][c] = blin[c] + sum_k (sums[g][k]/max(cnt[g],1)) * Wlin[k][c]
__global__ void k_head(const float* __restrict__ sums,
                       const float* __restrict__ cnt,
                       const float* __restrict__ Wlin,
                       const float* __restrict__ blin,
                       float* __restrict__ out, int g_total)
{
    int idx = blockIdx.x * blockDim.x + threadIdx.x;
    if (idx >= g_total * NCLS) return;
    const int g = idx / NCLS;
    const int c = idx % NCLS;
    const float inv = 1.0f / fmaxf(cnt[g], 1.0f);
    const float* sp = sums + (long long)g * EMB;
    float acc = 0.0f;
#pragma unroll 4
    for (int k = 0; k < EMB; ++k) acc += sp[k] * Wlin[k * NCLS + c];
    out[idx] = acc * inv + blin[c];
}

// ---------------------------------------------------------------------------
static inline size_t align256(size_t x) { return (x + 255) & ~(size_t)255; }

extern "C" void kernel_launch(void* const* d_in, const int* in_sizes, int n_in,
                              void* d_out, int out_size, void* d_ws, size_t ws_size,
                              hipStream_t stream)
{
    const int*   x        = (const int*)  d_in[0];
    const int*   eidx     = (const int*)  d_in[1];
    const int*   batch    = (const int*)  d_in[2];
    const float* shapeTab = (const float*)d_in[3];
    const float* colorTab = (const float*)d_in[4];
    const float* posTab   = (const float*)d_in[5];
    const float* W1       = (const float*)d_in[6];
    const float* b1       = (const float*)d_in[7];
    const float* W2       = (const float*)d_in[8];
    const float* b2       = (const float*)d_in[9];
    const float* Wlin     = (const float*)d_in[10];
    const float* blin     = (const float*)d_in[11];
    float*       out      = (float*)d_out;

    const int N = in_sizes[0] / 3;     // 100000
    const int E = in_sizes[1] / 2;     // 1600000
    const int G = out_size / NCLS;     // 1024

    const int* src = eidx;
    const int* dst = eidx + E;

    // workspace layout
    char* ws = (char*)d_ws;
    size_t off = 0;
    float* h    = (float*)(ws + off); off = align256(off + (size_t)N * EMB * 4);
    float* xw   = (float*)(ws + off); off = align256(off + (size_t)N * EMB * 4);
    float* agg  = (float*)(ws + off); off = align256(off + (size_t)N * EMB * 4);
    float* dinv = (float*)(ws + off); off = align256(off + (size_t)N * 4);
    float* sums = (float*)(ws + off); off = align256(off + (size_t)G * EMB * 4);
    float* cnt  = (float*)(ws + off); off = align256(off + (size_t)G * 4);
    (void)ws_size;

    const int TB = 256;
    const int nVec      = N * 32;                      // float4 units per node array
    const int gVecBlks  = (nVec + TB - 1) / TB;
    const int gEdgeBlks = (E * 32 + TB - 1) / TB;      // 32 lanes per edge
    const int gGemmBlks = (N + 127) / 128;

    // 1) embeddings
    k_embed<<<gVecBlks, TB, 0, stream>>>(x, shapeTab, colorTab, posTab, h, N);

    // 2) degrees (shared by both layers): deg = 1 (self loop) + in-degree; dinv = rsqrt
    k_fill<<<(N + TB - 1) / TB, TB, 0, stream>>>(dinv, 1.0f, N);
    k_deg_accum<<<(E + TB - 1) / TB, TB, 0, stream>>>(dst, dinv, E);
    k_rsqrt_inplace<<<(N + TB - 1) / TB, TB, 0, stream>>>(dinv, N);

    // 3) layer 1
    k_gemm_wmma<<<gGemmBlks, TB, 0, stream>>>(h, W1, xw, N);
    k_fill<<<4096, TB, 0, stream>>>(agg, 0.0f, N * EMB);
    k_agg_edges<<<gEdgeBlks, TB, 0, stream>>>(xw, src, dst, dinv, agg, E);
    k_finalize_relu<<<gVecBlks, TB, 0, stream>>>(agg, xw, dinv, b1, h, N);

    // 4) layer 2
    k_gemm_wmma<<<gGemmBlks, TB, 0, stream>>>(h, W2, xw, N);
    k_fill<<<4096, TB, 0, stream>>>(agg, 0.0f, N * EMB);
    k_agg_edges<<<gEdgeBlks, TB, 0, stream>>>(xw, src, dst, dinv, agg, E);
    k_finalize_relu<<<gVecBlks, TB, 0, stream>>>(agg, xw, dinv, b2, h, N);

    // 5) global mean pool + linear head
    k_fill<<<(G * EMB + TB - 1) / TB, TB, 0, stream>>>(sums, 0.0f, G * EMB);
    k_fill<<<(G + TB - 1) / TB, TB, 0, stream>>>(cnt, 0.0f, G);
    k_pool<<<gVecBlks, TB, 0, stream>>>(h, batch, sums, cnt, N);
    k_head<<<(G * NCLS + TB - 1) / TB, TB, 0, stream>>>(sums, cnt, Wlin, blin, out, G);
}